// GNN_3547642987348
// MI455X (gfx1250) — compile-verified
//
#include <hip/hip_runtime.h>
#include <hip/hip_bf16.h>

#define NN 50000      // nodes
#define NE 800000     // edges
#define D  128        // feature dim (D_IN == D_HID)
#define NG 512        // graphs
#define DOUT 2

typedef __attribute__((ext_vector_type(2))) float v2f;
typedef __attribute__((ext_vector_type(8))) float v8f;

// ---------------------------------------------------------------------------
// Degree / normalization:  deg[i] = 1 (self loop) + in-degree;  dis = rsqrt(deg)
// ---------------------------------------------------------------------------
__global__ void gcn_deg_init(float* __restrict__ deg) {
    int i = blockIdx.x * blockDim.x + threadIdx.x;
    if (i < NN) deg[i] = 1.0f;
}

__global__ void gcn_deg_count(const int* __restrict__ dst, float* __restrict__ deg) {
    int e = blockIdx.x * blockDim.x + threadIdx.x;   // grid covers NE exactly
    atomicAdd(&deg[dst[e]], 1.0f);
}

__global__ void gcn_dis_finalize(float* __restrict__ deg_to_dis) {
    int i = blockIdx.x * blockDim.x + threadIdx.x;
    if (i < NN) {
        float d = deg_to_dis[i];              // always >= 1
        deg_to_dis[i] = 1.0f / sqrtf(d);
    }
}

// ---------------------------------------------------------------------------
// Dense transform H = X @ W  (M x 128 @ 128 x 128) with V_WMMA_F32_16X16X4_F32.
// One wave per 16x64 output tile; A operand shared across 4 column sub-tiles.
// grid = (M/16, 128/64), block = 32 (one wave, no divergence -> EXEC all 1s).
// ---------------------------------------------------------------------------
__global__ __launch_bounds__(32) void gcn_gemm_wmma(
    const float* __restrict__ X, const float* __restrict__ W,
    float* __restrict__ H)
{
    const int rowBase = blockIdx.x * 16;
    const int colBase = blockIdx.y * 64;
    const int lane = threadIdx.x;
    const int half = lane >> 4;      // 0: lanes 0-15, 1: lanes 16-31
    const int l16  = lane & 15;

    v8f acc0 = {}, acc1 = {}, acc2 = {}, acc3 = {};

    const float* Xrow = X + (size_t)(rowBase + l16) * D;  // A: row M = l16
    const float* Wp   = W + colBase + l16;                // B: column base

    for (int k0 = 0; k0 < D; k0 += 4) {
        const int ka = k0 + half * 2;   // lanes 0-15: K=k0,k0+1 ; 16-31: K=k0+2,k0+3
        v2f a;
        a.x = Xrow[ka + 0];
        a.y = Xrow[ka + 1];

        const float* w0 = Wp + (size_t)ka * D;   // B row K=ka   at this column
        const float* w1 = w0 + D;                // B row K=ka+1

        v2f b;
        b.x = w0[0];  b.y = w1[0];
        acc0 = __builtin_amdgcn_wmma_f32_16x16x4_f32(false, a, false, b, (short)0, acc0, false, false);
        b.x = w0[16]; b.y = w1[16];
        acc1 = __builtin_amdgcn_wmma_f32_16x16x4_f32(false, a, false, b, (short)0, acc1, false, false);
        b.x = w0[32]; b.y = w1[32];
        acc2 = __builtin_amdgcn_wmma_f32_16x16x4_f32(false, a, false, b, (short)0, acc2, false, false);
        b.x = w0[48]; b.y = w1[48];
        acc3 = __builtin_amdgcn_wmma_f32_16x16x4_f32(false, a, false, b, (short)0, acc3, false, false);
    }

    // C/D layout: VGPR r -> (M = r + 8*half, N = l16) within the tile
    float* Hp = H + (size_t)(rowBase + half * 8) * D + colBase + l16;
#pragma unroll
    for (int r = 0; r < 8; ++r) {
        Hp[(size_t)r * D +  0] = acc0[r];
        Hp[(size_t)r * D + 16] = acc1[r];
        Hp[(size_t)r * D + 32] = acc2[r];
        Hp[(size_t)r * D + 48] = acc3[r];
    }
}

// ---------------------------------------------------------------------------
// Self-loop init: AGG[i][:] = Hlin[i][:] * dis[i]^2   (covers every element,
// so no memset of the 25.6MB accumulator is needed)
// ---------------------------------------------------------------------------
__global__ void gcn_agg_init(const float* __restrict__ Hlin,
                             const float* __restrict__ dis,
                             float* __restrict__ AGG)
{
    int idx = blockIdx.x * blockDim.x + threadIdx.x;   // over NN*32 float4s
    if (idx >= NN * (D / 4)) return;
    int node = idx >> 5;
    float di = dis[node];
    float nrm = di * di;
    float4 h = ((const float4*)Hlin)[idx];
    h.x *= nrm; h.y *= nrm; h.z *= nrm; h.w *= nrm;
    ((float4*)AGG)[idx] = h;
}

// ---------------------------------------------------------------------------
// Edge scatter: one wave per edge, 32 lanes x float4 = 128 features.
// AGG[dst] += Hlin[src] * dis[src]*dis[dst]
// ---------------------------------------------------------------------------
__global__ void gcn_scatter(const float* __restrict__ Hlin,
                            const int* __restrict__ src,
                            const int* __restrict__ dst,
                            const float* __restrict__ dis,
                            float* __restrict__ AGG)
{
    int e = blockIdx.x * (blockDim.x >> 5) + (threadIdx.x >> 5);
    if (e >= NE) return;
    int lane = threadIdx.x & 31;
    int s = src[e];
    int d = dst[e];
    float nrm = dis[s] * dis[d];
    float4 m = ((const float4*)(Hlin + (size_t)s * D))[lane];
    m.x *= nrm; m.y *= nrm; m.z *= nrm; m.w *= nrm;
    float* o = AGG + (size_t)d * D + lane * 4;
    atomicAdd(o + 0, m.x);
    atomicAdd(o + 1, m.y);
    atomicAdd(o + 2, m.z);
    atomicAdd(o + 3, m.w);
}

// ---------------------------------------------------------------------------
// In-place bias + ReLU:  H[i][c] = max(H[i][c] + b[c], 0)
// ---------------------------------------------------------------------------
__global__ void gcn_bias_relu(float* __restrict__ H, const float* __restrict__ b)
{
    int idx = blockIdx.x * blockDim.x + threadIdx.x;   // over NN*D
    if (idx >= NN * D) return;
    float v = H[idx] + b[idx & (D - 1)];
    H[idx] = v > 0.0f ? v : 0.0f;
}

// ---------------------------------------------------------------------------
// Pooling
// ---------------------------------------------------------------------------
__global__ void gcn_pool_zero(float* __restrict__ sums, float* __restrict__ cnts)
{
    int i = blockIdx.x * blockDim.x + threadIdx.x;
    if (i < NG * D) sums[i] = 0.0f;
    if (i < NG)     cnts[i] = 0.0f;
}

__global__ void gcn_pool_count(const int* __restrict__ batch, float* __restrict__ cnts)
{
    int i = blockIdx.x * blockDim.x + threadIdx.x;
    if (i < NN) atomicAdd(&cnts[batch[i]], 1.0f);
}

__global__ void gcn_pool_sum(const float* __restrict__ H,
                             const int* __restrict__ batch,
                             float* __restrict__ sums)
{
    int idx = blockIdx.x * blockDim.x + threadIdx.x;   // over NN*D
    if (idx >= NN * D) return;
    int g = batch[idx >> 7];
    atomicAdd(&sums[(size_t)g * D + (idx & (D - 1))], H[idx]);
}

// ---------------------------------------------------------------------------
// Classifier: out[g][o] = (sums[g]/max(cnt,1)) . Wc[:,o] + bc[o]
// ---------------------------------------------------------------------------
__global__ void gcn_classifier(const float* __restrict__ sums,
                               const float* __restrict__ cnts,
                               const float* __restrict__ Wc,
                               const float* __restrict__ bc,
                               float* __restrict__ out)
{
    int t = blockIdx.x * blockDim.x + threadIdx.x;
    if (t >= NG * DOUT) return;
    int g = t >> 1;
    int o = t & 1;
    float inv = 1.0f / fmaxf(cnts[g], 1.0f);
    const float* s = sums + (size_t)g * D;
    float acc = 0.0f;
    for (int c = 0; c < D; ++c) acc += s[c] * Wc[c * DOUT + o];
    out[t] = acc * inv + bc[o];
}

// ---------------------------------------------------------------------------
extern "C" void kernel_launch(void* const* d_in, const int* in_sizes, int n_in,
                              void* d_out, int out_size, void* d_ws, size_t ws_size,
                              hipStream_t stream)
{
    const float* x     = (const float*)d_in[0];
    const int*   ei    = (const int*)  d_in[1];   // (2, NE): row0 = src, row1 = dst
    const int*   batch = (const int*)  d_in[2];
    const float* W1    = (const float*)d_in[3];
    const float* b1    = (const float*)d_in[4];
    const float* W2    = (const float*)d_in[5];
    const float* b2    = (const float*)d_in[6];
    const float* Wc    = (const float*)d_in[7];
    const float* bc    = (const float*)d_in[8];
    float*       out   = (float*)d_out;

    const int* src = ei;
    const int* dst = ei + NE;

    // workspace layout (floats); offsets kept 16-float aligned
    float* ws   = (float*)d_ws;
    float* dis  = ws;                           // NN (holds deg then dis in place)
    float* bufA = ws + 50176;                   // NN*D  linear-transform output
    float* bufB = bufA + (size_t)NN * D;        // NN*D  aggregation / activations
    float* sums = bufB + (size_t)NN * D;        // NG*D
    float* cnts = sums + NG * D;                // NG

    const dim3 gGemm(NN / 16, D / 64);          // 3125 x 2 waves
    const int nodeBlocks  = (NN + 255) / 256;            // 196
    const int edgeBlocks  = NE / 256;                    // 3125 (exact)
    const int scatBlocks  = (NE + 7) / 8;                // 8 edges per 256-thr block
    const int elemBlocks  = (NN * D + 255) / 256;        // 25000
    const int vec4Blocks  = (NN * (D / 4) + 255) / 256;  // 6250
    const int poolZBlocks = (NG * D + 255) / 256;        // 256

    // structure: degree -> dis
    gcn_deg_init    <<<nodeBlocks, 256, 0, stream>>>(dis);
    gcn_deg_count   <<<edgeBlocks, 256, 0, stream>>>(dst, dis);
    gcn_dis_finalize<<<nodeBlocks, 256, 0, stream>>>(dis);

    // layer 1
    gcn_gemm_wmma   <<<gGemm, 32, 0, stream>>>(x, W1, bufA);
    gcn_agg_init    <<<vec4Blocks, 256, 0, stream>>>(bufA, dis, bufB);
    gcn_scatter     <<<scatBlocks, 256, 0, stream>>>(bufA, src, dst, dis, bufB);
    gcn_bias_relu   <<<elemBlocks, 256, 0, stream>>>(bufB, b1);

    // layer 2
    gcn_gemm_wmma   <<<gGemm, 32, 0, stream>>>(bufB, W2, bufA);
    gcn_agg_init    <<<vec4Blocks, 256, 0, stream>>>(bufA, dis, bufB);
    gcn_scatter     <<<scatBlocks, 256, 0, stream>>>(bufA, src, dst, dis, bufB);
    gcn_bias_relu   <<<elemBlocks, 256, 0, stream>>>(bufB, b2);

    // pool + classify
    gcn_pool_zero   <<<poolZBlocks, 256, 0, stream>>>(sums, cnts);
    gcn_pool_count  <<<nodeBlocks, 256, 0, stream>>>(batch, cnts);
    gcn_pool_sum    <<<elemBlocks, 256, 0, stream>>>(bufB, batch, sums);
    gcn_classifier  <<<(NG * DOUT + 255) / 256, 256, 0, stream>>>(sums, cnts, Wc, bc, out);
}